// StackedTransDecoder_5085241279167
// MI455X (gfx1250) — compile-verified
//
#include <hip/hip_runtime.h>
#include <math.h>

// ---------------------------------------------------------------------------
// Problem constants (match reference)
// ---------------------------------------------------------------------------
#define Bn   64
#define LXn  64
#define LYn  64
#define Dn   512
#define NEn  512
#define Hn   8
#define E2n  1024
#define D3n  1536

typedef float v2f __attribute__((ext_vector_type(2)));
typedef float v8f __attribute__((ext_vector_type(8)));

// ---------------------------------------------------------------------------
// fp32 WMMA GEMM: C[M,N] (+)= A[M,K] @ B[K,N]    (row-major, lda=K, ldb=N, ldc=N)
// One wave -> one 16x32 tile (two accumulators) via V_WMMA_F32_16X16X4_F32:
//  - A fragment loaded once per K-step, shared by both N-subtiles
//  - two independent WMMA accumulation chains for better pipelining
// Block = 256 threads = 8 waves covering 256 columns; grid = (N/256, M/16).
// A 16x4 layout: lanes 0-15 hold rows 0-15, VGPR0/1 = K {0,1} (half 0) / {2,3} (half 1).
// B 4x16 layout mirrors A with N striped across lanes.
// C/D: VGPR j, lanes 0-15 -> M=j, lanes 16-31 -> M=j+8.
// ---------------------------------------------------------------------------
template <bool ACCUM, bool TANH>
__global__ void wmma_gemm_f32(const float* __restrict__ A,
                              const float* __restrict__ B,
                              float* __restrict__ C,
                              int M, int N, int K,
                              const float* __restrict__ bias,      // length N or null
                              const float* __restrict__ rowscale,  // per-M scale or null
                              int rs_stride)
{
    const int wave  = threadIdx.x >> 5;
    const int lane  = threadIdx.x & 31;
    const int half  = lane >> 4;        // 0 or 1
    const int l16   = lane & 15;
    const int mBase = blockIdx.y * 16;
    const int nBase = blockIdx.x * 256 + wave * 32;
    if (nBase >= N || mBase >= M) return;

    v8f acc0 = {0.f, 0.f, 0.f, 0.f, 0.f, 0.f, 0.f, 0.f};
    v8f acc1 = acc0;
    if (ACCUM) {
        const float* cp = C + (mBase + half * 8) * N + nBase + l16;
#pragma unroll
        for (int j = 0; j < 8; ++j) {
            acc0[j] = cp[j * N];
            acc1[j] = cp[j * N + 16];
        }
    }

    // Pointer-bumped operand streams (32-bit strides, no per-iter index math).
    const float* ap  = A + (mBase + l16) * K + 2 * half;
    const float* bp0 = B + (2 * half) * N + nBase + l16;
    const float* bp1 = bp0 + 16;
    const int bstep = 4 * N;

#pragma unroll 4
    for (int k = 0; k < K; k += 4) {
        v2f a, b0, b1;
        a.x  = ap[0];
        a.y  = ap[1];
        b0.x = bp0[0];
        b0.y = bp0[N];
        b1.x = bp1[0];
        b1.y = bp1[N];
        __builtin_prefetch(bp0 + 16 * N);   // B panel, 16 k-steps ahead (WGP$)
        acc0 = __builtin_amdgcn_wmma_f32_16x16x4_f32(false, a, false, b0,
                                                     (short)0, acc0, false, false);
        acc1 = __builtin_amdgcn_wmma_f32_16x16x4_f32(false, a, false, b1,
                                                     (short)0, acc1, false, false);
        ap  += 4;
        bp0 += bstep;
        bp1 += bstep;
    }

    const float bv0 = bias ? bias[nBase + l16]      : 0.f;
    const float bv1 = bias ? bias[nBase + 16 + l16] : 0.f;
    float* cp = C + (mBase + half * 8) * N + nBase + l16;
    const float* rp = rowscale ? rowscale + (size_t)(mBase + half * 8) * rs_stride : nullptr;
#pragma unroll
    for (int j = 0; j < 8; ++j) {
        float v0 = acc0[j] + bv0;
        float v1 = acc1[j] + bv1;
        if (TANH) { v0 = tanhf(v0); v1 = tanhf(v1); }
        if (rowscale) {
            const float rs = rp[(size_t)j * rs_stride];
            v0 *= rs; v1 *= rs;
        }
        cp[j * N]      = v0;
        cp[j * N + 16] = v1;
    }
}

// ---------------------------------------------------------------------------
// Block-wide sum reduction (256 threads)
// ---------------------------------------------------------------------------
__device__ inline float block_sum(float v, float* red)
{
    const int t = threadIdx.x;
    red[t] = v;
    __syncthreads();
    for (int s = 128; s > 0; s >>= 1) {
        if (t < s) red[t] += red[t + s];
        __syncthreads();
    }
    const float r = red[0];
    __syncthreads();
    return r;
}

__device__ inline float sigmoidf(float x) { return 1.f / (1.f + expf(-x)); }

// ---------------------------------------------------------------------------
// Masked mean-pool of annotations: pooled[b,e] = sum_l xs_h*mask / sum_l mask
// ---------------------------------------------------------------------------
__global__ void pool_kernel(const float* __restrict__ xs_h,
                            const float* __restrict__ xs_mask,
                            float* __restrict__ pooled)
{
    const int b = blockIdx.x;
    float msum = 0.f;
#pragma unroll
    for (int l = 0; l < LXn; ++l) msum += xs_mask[b * LXn + l];
    const float inv = 1.f / msum;
    for (int e = threadIdx.x; e < E2n; e += 256) {
        float s = 0.f;
        for (int l = 0; l < LXn; ++l)
            s += xs_h[((size_t)b * LXn + l) * E2n + e] * xs_mask[b * LXn + l];
        pooled[(size_t)b * E2n + e] = s * inv;
    }
}

// ---------------------------------------------------------------------------
// In-place LayerNorm over 512-wide gate chunks: data row r -> chunk
// (r/3)*1536 + (r%3)*512 ; out = norm * g[gate] + bvec[gate]
// ---------------------------------------------------------------------------
__global__ void ln_scale_rows(float* __restrict__ data,
                              const float* __restrict__ g,
                              const float* __restrict__ bvec)
{
    __shared__ float red[256];
    const int r = blockIdx.x, t = threadIdx.x;
    const int gate = r % 3;
    float* row = data + (size_t)(r / 3) * D3n + gate * Dn;
    const float v0 = row[t], v1 = row[t + 256];
    const float mu = block_sum(v0 + v1, red) * (1.f / (float)Dn);
    const float d0 = v0 - mu, d1 = v1 - mu;
    const float var = block_sum(d0 * d0 + d1 * d1, red) * (1.f / (float)Dn);
    const float rstd = rsqrtf(var + 1e-5f);
    row[t]       = d0 * rstd * g[gate * Dn + t]       + bvec[gate * Dn + t];
    row[t + 256] = d1 * rstd * g[gate * Dn + t + 256] + bvec[gate * Dn + t + 256];
}

// ---------------------------------------------------------------------------
// LayerNorm-GRU combine. One block per batch row.
//  NORM_AX=false: ax already = LN(x@Wx)*gx+bx (precomputed), stride ax_row_stride.
//  NORM_AX=true : ax raw, LN'd here with gx/bx.
//  ah always raw (h@Wh), LN'd with gh (no bias).
// ---------------------------------------------------------------------------
template <bool NORM_AX>
__global__ void ln_gru_kernel(const float* __restrict__ ax, int ax_row_stride,
                              const float* __restrict__ gx, const float* __restrict__ bx,
                              const float* __restrict__ ah,
                              const float* __restrict__ gh,
                              const float* __restrict__ hprev,
                              const float* __restrict__ mask, int mask_stride,
                              float* __restrict__ hout)
{
    __shared__ float red[256];
    __shared__ float shn[3 * Dn];
    __shared__ float sxn[3 * Dn];
    const int b = blockIdx.x, t = threadIdx.x;
    const float* axr = ax + (size_t)b * ax_row_stride;
    const float* ahr = ah + (size_t)b * D3n;

    for (int g = 0; g < 3; ++g) {
        {   // LN of h-path gate chunk
            const float v0 = ahr[g * Dn + t], v1 = ahr[g * Dn + t + 256];
            const float mu = block_sum(v0 + v1, red) * (1.f / (float)Dn);
            const float d0 = v0 - mu, d1 = v1 - mu;
            const float var = block_sum(d0 * d0 + d1 * d1, red) * (1.f / (float)Dn);
            const float rstd = rsqrtf(var + 1e-5f);
            shn[g * Dn + t]       = d0 * rstd * gh[g * Dn + t];
            shn[g * Dn + t + 256] = d1 * rstd * gh[g * Dn + t + 256];
        }
        if (NORM_AX) {   // LN of x-path gate chunk
            const float v0 = axr[g * Dn + t], v1 = axr[g * Dn + t + 256];
            const float mu = block_sum(v0 + v1, red) * (1.f / (float)Dn);
            const float d0 = v0 - mu, d1 = v1 - mu;
            const float var = block_sum(d0 * d0 + d1 * d1, red) * (1.f / (float)Dn);
            const float rstd = rsqrtf(var + 1e-5f);
            sxn[g * Dn + t]       = d0 * rstd * gx[g * Dn + t]       + bx[g * Dn + t];
            sxn[g * Dn + t + 256] = d1 * rstd * gx[g * Dn + t + 256] + bx[g * Dn + t + 256];
        }
    }
    __syncthreads();

    const float m = mask[(size_t)b * mask_stride];
    for (int d = t; d < Dn; d += 256) {
        const float ax0 = NORM_AX ? sxn[d]          : axr[d];
        const float ax1 = NORM_AX ? sxn[Dn + d]     : axr[Dn + d];
        const float ax2 = NORM_AX ? sxn[2 * Dn + d] : axr[2 * Dn + d];
        const float r  = sigmoidf(ax0 + shn[d]);
        const float z  = sigmoidf(ax1 + shn[Dn + d]);
        const float hc = tanhf(ax2 + r * shn[2 * Dn + d]);
        const float h  = hprev[(size_t)b * Dn + d];
        const float hn = (1.f - z) * h + z * hc;
        hout[(size_t)b * Dn + d] = m * hn + (1.f - m) * h;
    }
}

// ---------------------------------------------------------------------------
// Multihead additive attention, one block per batch row (256 threads).
//  e[l,h] = sum_d tanh(uh[b,l,d] + q[b,d]) * V[d,h]; softmax over l;
//  ctx[b, h*128+v] = sum_l alpha[l,h] * xs_h[b,l,h*128+v]
// ---------------------------------------------------------------------------
__global__ void attention_kernel(const float* __restrict__ uh,
                                 const float* __restrict__ q,
                                 const float* __restrict__ xs_h,
                                 const float* __restrict__ xs_mask,
                                 const float* __restrict__ V,
                                 float* __restrict__ ctx_raw)
{
    __shared__ float qs[Dn];
    __shared__ float vs[Dn * Hn];
    __shared__ float part[LXn * 4 * Hn];
    __shared__ float eh[LXn * Hn];
    __shared__ float sinv[Hn];
    const int b = blockIdx.x, t = threadIdx.x;

    for (int i = t; i < Dn; i += 256) qs[i] = q[(size_t)b * Dn + i];
    for (int i = t; i < Dn * Hn; i += 256) vs[i] = V[i];
    __syncthreads();

    {   // per-head partial additive scores over a 128-wide d-chunk
        const int l = t >> 2, c = t & 3, d0 = c * 128;
        float acc[Hn];
#pragma unroll
        for (int h = 0; h < Hn; ++h) acc[h] = 0.f;
        const float* uhr = uh + ((size_t)b * LXn + l) * Dn;
        for (int d = d0; d < d0 + 128; ++d) {
            const float hv = tanhf(uhr[d] + qs[d]);
#pragma unroll
            for (int h = 0; h < Hn; ++h) acc[h] += hv * vs[d * Hn + h];
        }
#pragma unroll
        for (int h = 0; h < Hn; ++h) part[(l * 4 + c) * Hn + h] = acc[h];
    }
    __syncthreads();

    for (int i = t; i < LXn * Hn; i += 256) {   // fold partials + mask
        const int l = i >> 3, h = i & 7;
        float e = part[(l * 4 + 0) * Hn + h] + part[(l * 4 + 1) * Hn + h] +
                  part[(l * 4 + 2) * Hn + h] + part[(l * 4 + 3) * Hn + h];
        if (!(xs_mask[b * LXn + l] > 0.f)) e = -1e9f;
        eh[l * Hn + h] = e;
    }
    __syncthreads();

    if (t < Hn) {   // softmax over l, per head
        float mx = -1e30f;
        for (int l = 0; l < LXn; ++l) mx = fmaxf(mx, eh[l * Hn + t]);
        float sum = 0.f;
        for (int l = 0; l < LXn; ++l) {
            const float a = expf(eh[l * Hn + t] - mx);
            eh[l * Hn + t] = a;
            sum += a;
        }
        sinv[t] = 1.f / sum;
    }
    __syncthreads();
    for (int i = t; i < LXn * Hn; i += 256) eh[i] *= sinv[i & 7];
    __syncthreads();

    for (int o = t; o < E2n; o += 256) {   // head-sliced value gather
        const int h = o >> 7, v = o & 127;
        float s = 0.f;
        for (int l = 0; l < LXn; ++l)
            s += eh[l * Hn + h] * xs_h[((size_t)b * LXn + l) * E2n + h * 128 + v];
        ctx_raw[(size_t)b * E2n + o] = s;
    }
}

// ---------------------------------------------------------------------------
// Final logit: out[b,t,:] = tanh(tmp + yWy_all[b,t,:] + by+bc+bs) * ys_mask[b,t]
// ---------------------------------------------------------------------------
__global__ void logit_kernel(const float* __restrict__ tmp,
                             const float* __restrict__ yWy,
                             const float* __restrict__ by,
                             const float* __restrict__ bc,
                             const float* __restrict__ bs,
                             const float* __restrict__ ys_mask,
                             int t, float* __restrict__ out)
{
    const int i = blockIdx.x * 256 + threadIdx.x;   // 0 .. B*D-1
    const int b = i >> 9, d = i & 511;
    const size_t o = ((size_t)b * LYn + t) * Dn + d;
    const float m = ys_mask[b * LYn + t];
    out[o] = tanhf(tmp[i] + yWy[o] + by[d] + bc[d] + bs[d]) * m;
}

// ---------------------------------------------------------------------------
// Host-side launcher
// ---------------------------------------------------------------------------
static void gemm(const float* A, const float* B, float* C, int M, int N, int K,
                 const float* bias, const float* rs, int rs_stride,
                 bool accum, bool tanh_ep, hipStream_t stream)
{
    dim3 grid(N / 256, M / 16), block(256);
    if (accum)
        wmma_gemm_f32<true, false><<<grid, block, 0, stream>>>(A, B, C, M, N, K, bias, rs, rs_stride);
    else if (tanh_ep)
        wmma_gemm_f32<false, true><<<grid, block, 0, stream>>>(A, B, C, M, N, K, bias, rs, rs_stride);
    else
        wmma_gemm_f32<false, false><<<grid, block, 0, stream>>>(A, B, C, M, N, K, bias, rs, rs_stride);
}

extern "C" void kernel_launch(void* const* d_in, const int* in_sizes, int n_in,
                              void* d_out, int out_size, void* d_ws, size_t ws_size,
                              hipStream_t stream)
{
    const float* xs_h    = (const float*)d_in[0];
    const float* ys_e    = (const float*)d_in[1];
    const float* xs_mask = (const float*)d_in[2];
    const float* ys_mask = (const float*)d_in[3];
    const float* W_sinit = (const float*)d_in[4];
    const float* b_sinit = (const float*)d_in[5];
    const float* W_keys  = (const float*)d_in[6];
    const float* Wx_cell = (const float*)d_in[7];
    const float* bx_cell = (const float*)d_in[8];
    const float* gx_cell = (const float*)d_in[9];
    const float* Wh_cell = (const float*)d_in[10];
    const float* gh_cell = (const float*)d_in[11];
    const float* Wx_cond = (const float*)d_in[12];
    const float* bx_cond = (const float*)d_in[13];
    const float* gx_cond = (const float*)d_in[14];
    const float* Wh_cond = (const float*)d_in[15];
    const float* gh_cond = (const float*)d_in[16];
    const float* Wq      = (const float*)d_in[17];
    const float* bq      = (const float*)d_in[18];
    const float* V_att   = (const float*)d_in[19];
    const float* Wo      = (const float*)d_in[20];
    const float* bo      = (const float*)d_in[21];
    const float* Wy      = (const float*)d_in[22];
    const float* by      = (const float*)d_in[23];
    const float* Wc      = (const float*)d_in[24];
    const float* bc      = (const float*)d_in[25];
    const float* Ws      = (const float*)d_in[26];
    const float* bs      = (const float*)d_in[27];
    float* out = (float*)d_out;

    // ---- workspace layout (floats) ----
    float* w = (float*)d_ws;
    size_t off = 0;
    float* uh      = w + off; off += (size_t)Bn * LXn * Dn;        // 2,097,152
    float* ax_all  = w + off; off += (size_t)Bn * LYn * D3n;       // 6,291,456
    float* yWy_all = w + off; off += (size_t)Bn * LYn * Dn;        // 2,097,152
    float* pooled  = w + off; off += (size_t)Bn * E2n;
    float* sA      = w + off; off += (size_t)Bn * Dn;
    float* sB      = w + off; off += (size_t)Bn * Dn;
    float* state   = w + off; off += (size_t)Bn * Dn;
    float* ahbuf   = w + off; off += (size_t)Bn * D3n;
    float* axcbuf  = w + off; off += (size_t)Bn * D3n;
    float* qbuf    = w + off; off += (size_t)Bn * Dn;
    float* ctx_raw = w + off; off += (size_t)Bn * E2n;
    float* ctx2    = w + off; off += (size_t)Bn * E2n;
    float* tmpl    = w + off; off += (size_t)Bn * Dn;
    if (ws_size < off * sizeof(float)) return;   // insufficient scratch: bail safely

    const int BL = Bn * LYn;   // 4096 rows, time-minor (row = b*LY + t)

    // ---- one-shot precompute (all heavy, recurrence-independent GEMMs) ----
    // attention keys: uh = xs_h @ W_keys        (4096 x 512, K=1024)
    gemm(xs_h, W_keys, uh, Bn * LXn, Dn, E2n, nullptr, nullptr, 0, false, false, stream);
    // s0 = tanh(meanpool(xs_h) @ W_sinit + b)
    pool_kernel<<<Bn, 256, 0, stream>>>(xs_h, xs_mask, pooled);
    gemm(pooled, W_sinit, sA, Bn, Dn, E2n, b_sinit, nullptr, 0, false, true, stream);
    // x-path of cell GRU for all timesteps: LN(ys_e @ Wx_cell)*gx+bx
    gemm(ys_e, Wx_cell, ax_all, BL, D3n, NEn, nullptr, nullptr, 0, false, false, stream);
    ln_scale_rows<<<BL * 3, 256, 0, stream>>>(ax_all, gx_cell, bx_cell);
    // logit y-path for all timesteps: ys_e @ Wy
    gemm(ys_e, Wy, yWy_all, BL, Dn, NEn, nullptr, nullptr, 0, false, false, stream);

    // ---- sequential decode ----
    for (int t = 0; t < LYn; ++t) {
        const float* s_prev = (t & 1) ? sB : sA;
        float*       s_next = (t & 1) ? sA : sB;
        const float* mt = ys_mask + t;           // per-row stride LYn

        // cell GRU: ah = s_prev @ Wh_cell; combine with precomputed x-path
        gemm(s_prev, Wh_cell, ahbuf, Bn, D3n, Dn, nullptr, nullptr, 0, false, false, stream);
        ln_gru_kernel<false><<<Bn, 256, 0, stream>>>(ax_all + (size_t)t * D3n, LYn * D3n,
                                                     nullptr, nullptr, ahbuf, gh_cell,
                                                     s_prev, mt, LYn, state);
        // attention
        gemm(state, Wq, qbuf, Bn, Dn, Dn, bq, nullptr, 0, false, false, stream);
        attention_kernel<<<Bn, 256, 0, stream>>>(uh, qbuf, xs_h, xs_mask, V_att, ctx_raw);
        gemm(ctx_raw, Wo, ctx2, Bn, E2n, E2n, bo, mt, LYn, false, false, stream);  // *ym
        // cond GRU
        gemm(ctx2, Wx_cond, axcbuf, Bn, D3n, E2n, nullptr, nullptr, 0, false, false, stream);
        gemm(state, Wh_cond, ahbuf, Bn, D3n, Dn, nullptr, nullptr, 0, false, false, stream);
        ln_gru_kernel<true><<<Bn, 256, 0, stream>>>(axcbuf, D3n, gx_cond, bx_cond,
                                                    ahbuf, gh_cond, state, mt, LYn, s_next);
        // logit = tanh(yWy + ctx2@Wc + s_t@Ws + biases) * ym
        gemm(ctx2, Wc, tmpl, Bn, Dn, E2n, nullptr, nullptr, 0, false, false, stream);
        gemm(s_next, Ws, tmpl, Bn, Dn, Dn, nullptr, nullptr, 0, true, false, stream);
        logit_kernel<<<(Bn * Dn) / 256, 256, 0, stream>>>(tmpl, yWy_all, by, bc, bs,
                                                          ys_mask, t, out);
    }
}